// SurvLoss_4621384810914
// MI455X (gfx1250) — compile-verified
//
#include <hip/hip_runtime.h>
#include <hip/hip_bf16.h>

typedef __attribute__((ext_vector_type(2))) float v2f;
typedef __attribute__((ext_vector_type(8))) float v8f;

#define NBINS 2048   // times are 1..2000; bin 0 unused, padded to 2048
#define P1_THREADS 256

// ---------------------------------------------------------------------------
// Pass 1: streaming histogram. Each block builds a private LDS histogram of
// P[t] = sum exp(Yhat) over events at time t, D[t] = event count at time t,
// plus scalar partials (negTotal, loss1, obs), then flushes its histogram to
// its own region of d_ws (plain stores -> deterministic structure).
// ---------------------------------------------------------------------------
__global__ __launch_bounds__(P1_THREADS) void surv_pass1(
    const float* __restrict__ Yhat, const float* __restrict__ Y,
    float* __restrict__ ws, int n, int n4, unsigned long long sbase, int nb) {
  __shared__ float hP[NBINS];
  __shared__ float hD[NBINS];
  __shared__ float red[P1_THREADS];

  for (int i = threadIdx.x; i < NBINS; i += P1_THREADS) { hP[i] = 0.f; hD[i] = 0.f; }
  __syncthreads();

  float negAcc = 0.f, l1 = 0.f, obs = 0.f;

  const float4* y4 = (const float4*)Y;
  const float4* h4 = (const float4*)Yhat;
  const int stride = gridDim.x * blockDim.x;

  for (int i = blockIdx.x * blockDim.x + threadIdx.x; i < n4; i += stride) {
    if (i + stride < n4) {            // pull next tile toward L2/WGP$
      __builtin_prefetch(&y4[i + stride], 0, 1);
      __builtin_prefetch(&h4[i + stride], 0, 1);
    }
    float4 yv = y4[i];
    float4 hv = h4[i];
    float ys[4] = {yv.x, yv.y, yv.z, yv.w};
    float hs[4] = {hv.x, hv.y, hv.z, hv.w};
#pragma unroll
    for (int c = 0; c < 4; ++c) {
      float e = __expf(hs[c]);                       // v_exp_f32 path
      unsigned t = (unsigned)(int)fabsf(ys[c]);
      t = t < (NBINS - 1) ? t : (NBINS - 1);
      if (ys[c] > 0.f) {
        atomicAdd(&hP[t], e);                        // ds_add_f32
        atomicAdd(&hD[t], 1.f);
        l1 += hs[c];
        obs += 1.f;
      } else {
        negAcc += e;
      }
    }
  }
  // scalar tail (N % 4), handled once
  if (blockIdx.x == 0 && threadIdx.x == 0) {
    for (int i = n4 * 4; i < n; ++i) {
      float yv = Y[i], hvv = Yhat[i];
      float e = __expf(hvv);
      unsigned t = (unsigned)(int)fabsf(yv);
      t = t < (NBINS - 1) ? t : (NBINS - 1);
      if (yv > 0.f) { atomicAdd(&hP[t], e); atomicAdd(&hD[t], 1.f); l1 += hvv; obs += 1.f; }
      else negAcc += e;
    }
  }
  __syncthreads();

  // flush private histogram: block r owns ws[r*4096 .. r*4096+4095]
  float* base = ws + (unsigned long long)blockIdx.x * (2 * NBINS);
  for (int i = threadIdx.x; i < NBINS; i += P1_THREADS) {
    base[i] = hP[i];
    base[NBINS + i] = hD[i];
  }

  // block-reduce the three scalars (tree; wave32-safe, syncthreads-based)
  red[threadIdx.x] = negAcc; __syncthreads();
  for (int s = P1_THREADS / 2; s > 0; s >>= 1) {
    if (threadIdx.x < s) red[threadIdx.x] += red[threadIdx.x + s];
    __syncthreads();
  }
  if (threadIdx.x == 0) ws[sbase + blockIdx.x] = red[0];
  __syncthreads();
  red[threadIdx.x] = l1; __syncthreads();
  for (int s = P1_THREADS / 2; s > 0; s >>= 1) {
    if (threadIdx.x < s) red[threadIdx.x] += red[threadIdx.x + s];
    __syncthreads();
  }
  if (threadIdx.x == 0) ws[sbase + nb + blockIdx.x] = red[0];
  __syncthreads();
  red[threadIdx.x] = obs; __syncthreads();
  for (int s = P1_THREADS / 2; s > 0; s >>= 1) {
    if (threadIdx.x < s) red[threadIdx.x] += red[threadIdx.x + s];
    __syncthreads();
  }
  if (threadIdx.x == 0) ws[sbase + 2 * nb + blockIdx.x] = red[0];
}

// ---------------------------------------------------------------------------
// Pass 2: cross-block reduction of the NB per-block histograms via WMMA.
// D = A x B + C with A = ones(16x4): every row of D is the column sum of the
// 4x16 tile of B, so chaining C over all block-rows yields, in D row 0
// (accumulator VGPR0, lanes 0..15), the total P[bin] / D[bin] for 16 bins.
// One wave per 16 bins; 128 waves cover 2048 bins.
// ---------------------------------------------------------------------------
__global__ __launch_bounds__(256) void surv_pass2(
    const float* __restrict__ ws, float* __restrict__ outPD, int nb) {
  int wave = blockIdx.x * (blockDim.x / 32) + (threadIdx.x >> 5);
  int lane = threadIdx.x & 31;
  int n0 = wave * 16;
  int col = n0 + (lane & 15);
  int half = lane >> 4;

  v8f cP = {};
  v8f cD = {};
  v2f a; a.x = 1.0f; a.y = 1.0f;   // A = ones(16x4): layout-independent

  for (int k0 = 0; k0 < nb; k0 += 4) {
    const float* r0 = ws + (unsigned long long)(k0 + half * 2) * (2 * NBINS);
    const float* r1 = ws + (unsigned long long)(k0 + half * 2 + 1) * (2 * NBINS);
    v2f bP; bP.x = r0[col];          bP.y = r1[col];
    v2f bD; bD.x = r0[NBINS + col];  bD.y = r1[NBINS + col];
    cP = __builtin_amdgcn_wmma_f32_16x16x4_f32(false, a, false, bP, (short)0, cP,
                                               false, false);
    cD = __builtin_amdgcn_wmma_f32_16x16x4_f32(false, a, false, bD, (short)0, cD,
                                               false, false);
  }
  // C/D layout: VGPR0, lanes 0..15 hold (M=0, N=lane) -> bin totals
  if (lane < 16) {
    outPD[n0 + lane] = cP[0];
    outPD[NBINS + n0 + lane] = cD[0];
  }
}

// ---------------------------------------------------------------------------
// Pass 3: one block. Reduce per-block scalars, inclusive-scan P over 2048
// bins in LDS (pair-compaction + Hillis-Steele over 1024), then
// loss2 = sum_t D[t]*log(negTotal + prefixP[t]); out = (loss2 - loss1)/obs.
// ---------------------------------------------------------------------------
__global__ __launch_bounds__(1024) void surv_pass3(
    const float* __restrict__ ws, int nb, unsigned long long sbase,
    unsigned long long rbase, float* __restrict__ out) {
  __shared__ float sP[NBINS];
  __shared__ float sQ[1024];
  __shared__ float sred[1024];
  __shared__ float sc[3];
  int tid = threadIdx.x;

  sP[tid] = ws[rbase + tid];
  sP[tid + 1024] = ws[rbase + tid + 1024];

  float neg = 0.f, l1 = 0.f, ob = 0.f;
  for (int r = tid; r < nb; r += 1024) {
    neg += ws[sbase + r];
    l1  += ws[sbase + nb + r];
    ob  += ws[sbase + 2 * nb + r];
  }
  sred[tid] = neg; __syncthreads();
  for (int s = 512; s > 0; s >>= 1) { if (tid < s) sred[tid] += sred[tid + s]; __syncthreads(); }
  if (tid == 0) sc[0] = sred[0];
  __syncthreads();
  sred[tid] = l1; __syncthreads();
  for (int s = 512; s > 0; s >>= 1) { if (tid < s) sred[tid] += sred[tid + s]; __syncthreads(); }
  if (tid == 0) sc[1] = sred[0];
  __syncthreads();
  sred[tid] = ob; __syncthreads();
  for (int s = 512; s > 0; s >>= 1) { if (tid < s) sred[tid] += sred[tid + s]; __syncthreads(); }
  if (tid == 0) sc[2] = sred[0];
  __syncthreads();

  // inclusive scan of sP[0..2047]: pair-sums then 1024-wide Hillis-Steele
  float q = sP[2 * tid] + sP[2 * tid + 1];
  sQ[tid] = q; __syncthreads();
  for (int off = 1; off < 1024; off <<= 1) {
    float v = (tid >= off) ? sQ[tid - off] : 0.f;
    __syncthreads();
    sQ[tid] += v;
    __syncthreads();
  }
  float Qi = sQ[tid];
  float Qe = (tid > 0) ? sQ[tid - 1] : 0.f;

  float negT = sc[0];
  float pre0 = negT + Qe + sP[2 * tid];   // inclusive prefix at bin 2*tid
  float pre1 = negT + Qi;                 // inclusive prefix at bin 2*tid+1
  float d0 = ws[rbase + NBINS + 2 * tid];
  float d1 = ws[rbase + NBINS + 2 * tid + 1];
  float c = 0.f;
  if (d0 > 0.f) c += d0 * logf(pre0);
  if (d1 > 0.f) c += d1 * logf(pre1);

  sred[tid] = c; __syncthreads();
  for (int s = 512; s > 0; s >>= 1) { if (tid < s) sred[tid] += sred[tid + s]; __syncthreads(); }
  if (tid == 0) out[0] = (sred[0] - sc[1]) / sc[2];
}

extern "C" void kernel_launch(void* const* d_in, const int* in_sizes, int n_in,
                              void* d_out, int out_size, void* d_ws, size_t ws_size,
                              hipStream_t stream) {
  const float* Yhat = (const float*)d_in[0];
  const float* Y    = (const float*)d_in[1];
  float* ws = (float*)d_ws;
  int n = in_sizes[0];
  int n4 = n / 4;

  // NB pass-1 blocks, each owning 2*2048 floats of scratch; NB multiple of 4
  // so the WMMA K-chunks in pass 2 never read unwritten scratch.
  size_t floatsAvail = ws_size / sizeof(float);
  int nb = 512;
  while (nb > 4 &&
         (size_t)nb * (2 * NBINS) + 3 * (size_t)nb + 2 * NBINS > floatsAvail)
    nb -= 4;
  unsigned long long sbase = (unsigned long long)nb * (2 * NBINS);
  unsigned long long rbase = sbase + 3ull * (unsigned long long)nb;

  surv_pass1<<<nb, P1_THREADS, 0, stream>>>(Yhat, Y, ws, n, n4, sbase, nb);
  surv_pass2<<<16, 256, 0, stream>>>(ws, ws + rbase, nb);   // 128 waves x 16 bins
  surv_pass3<<<1, 1024, 0, stream>>>(ws, nb, sbase, rbase, (float*)d_out);
}